// LoRA_qkv_1468878815513
// MI455X (gfx1250) — compile-verified
//
#include <hip/hip_runtime.h>

// LoRA-QKV fused projection for MI455X (gfx1250, wave32, WMMA + TDM).
//   out = concat(q, k, v),  qkv = X @ W^T + b
//   q  += (X @ Aq^T) @ Bq^T * (1/16),  v += (X @ Av^T) @ Bv^T * (1/16)
// X 16384x1024 f32, W 3072x1024, out 16384x3072. Compute-bound (AI~384 F/B).
// Main loop: V_WMMA_F32_16X16X4_F32 core fed by TENSOR_LOAD_TO_LDS (TDM DMA,
// double-buffered LDS, s_wait_tensorcnt). Epilogue: LoRA up-projection also
// done with WMMA (rank-16 = 4 K-steps), bias add, nontemporal streaming store.

#define D_MODEL 1024
#define N_OUT   3072
#define RANK    16
#define SCALING (1.0f / 16.0f)

#define BM 128
#define BN 128
#define BK 32
#define LDS_STRIDE 36   // 32 + 4 pad (TDM pad: interval=32 dwords, amount=4 dwords)

typedef float v2f __attribute__((ext_vector_type(2)));
typedef float v8f __attribute__((ext_vector_type(8)));
typedef unsigned int v4u __attribute__((ext_vector_type(4)));
typedef int v4i __attribute__((ext_vector_type(4)));
typedef int v8i __attribute__((ext_vector_type(8)));

// ---------------------------------------------------------------------------
// TDM 2D tile load: tile_h x tile_w f32 elements from a row-major tensor
// (row stride = stride_elems) into LDS at lds_addr, padding each 32-dword
// row with 4 dwords (-> 36-float LDS row stride, bank-conflict free).
// Descriptor layout per CDNA5 ISA.
// This toolchain's builtin is the 6-arg form.
// ---------------------------------------------------------------------------
__device__ __forceinline__ void tdm_load_tile_f32(
    unsigned int lds_addr, const float* gptr,
    unsigned int tile_w, unsigned int tile_h,
    unsigned int tensor_w, unsigned int tensor_h,
    unsigned int stride_elems) {
  unsigned long long ga = (unsigned long long)(uintptr_t)gptr;

  v4u g0;
  g0[0] = 1u;                                              // count=1 (valid), user mode
  g0[1] = lds_addr;                                        // lds_addr [63:32]
  g0[2] = (unsigned int)(ga & 0xFFFFFFFFu);                // global_addr [95:64]
  g0[3] = (unsigned int)((ga >> 32) & 0x01FFFFFFu)         // global_addr [120:96]
        | (2u << 30);                                      // type=2 ("image") [127:126]

  v8i g1;
  g1[0] = (int)((2u << 16)      // data_size = 4 bytes
              | (1u << 20)      // pad_enable
              | (4u << 22)      // pad_interval: code 4 = every 32 dwords
              | (3u << 25));    // pad_amount:   code 3 = 4 dwords
  g1[1] = (int)((tensor_w & 0xFFFFu) << 16);                       // tensor_dim0 lo16
  g1[2] = (int)(((tensor_w >> 16) & 0xFFFFu)
              | ((tensor_h & 0xFFFFu) << 16));                     // dim0 hi / dim1 lo
  g1[3] = (int)(((tensor_h >> 16) & 0xFFFFu)
              | ((tile_w & 0xFFFFu) << 16));                       // dim1 hi / tile_dim0
  g1[4] = (int)(tile_h & 0xFFFFu);                                 // tile_dim1 (tile_dim2=0)
  g1[5] = (int)stride_elems;                                       // tensor_dim0_stride lo32
  g1[6] = 0;                                                       // stride hi16, dim1_stride lo16
  g1[7] = 0;                                                       // dim1_stride hi32

  v4i g2 = {0, 0, 0, 0};   // 2D tensor: groups 2/3 unused
  v4i g3 = {0, 0, 0, 0};
  v8i g4 = {0, 0, 0, 0, 0, 0, 0, 0};

  __builtin_amdgcn_tensor_load_to_lds(g0, g1, g2, g3, g4, 0);
}

// ---------------------------------------------------------------------------
// Kernel 1: LoRA down-projection. T[m][0..15]=X[m]·Aq[r], T[m][16..31]=X[m]·Av[r]
// One wave32 per row m; lane r computes one rank's dot product.
// ---------------------------------------------------------------------------
__global__ __launch_bounds__(256)
void lora_down_kernel(const float* __restrict__ x,
                      const float* __restrict__ a_q,
                      const float* __restrict__ a_v,
                      float* __restrict__ T) {
    const int lane = threadIdx.x & 31;
    const int wave = threadIdx.x >> 5;
    const int m = blockIdx.x * 8 + wave;

    const float* __restrict__ xrow = x + (size_t)m * D_MODEL;
    const float* __restrict__ arow = (lane < 16)
        ? (a_q + (size_t)lane * D_MODEL)
        : (a_v + (size_t)(lane - 16) * D_MODEL);

    float acc = 0.0f;
    #pragma unroll 4
    for (int d = 0; d < D_MODEL; d += 4) {
        float4 xv = *(const float4*)(xrow + d);
        float4 av = *(const float4*)(arow + d);
        acc += xv.x * av.x + xv.y * av.y + xv.z * av.z + xv.w * av.w;
    }
    T[(size_t)m * 32 + lane] = acc;
}

// ---------------------------------------------------------------------------
// Kernel 2: WMMA GEMM, TDM-staged double-buffered LDS, WMMA LoRA epilogue.
// 256 threads = 8 waves (2M x 4N); wave tile 64x32 = 4x2 f32 WMMA accumulators.
// ---------------------------------------------------------------------------
__global__ __launch_bounds__(256)
void qkv_wmma_kernel(const float* __restrict__ x,
                     const float* __restrict__ w,
                     const float* __restrict__ bias,
                     const float* __restrict__ b_q,
                     const float* __restrict__ b_v,
                     const float* __restrict__ T,
                     float* __restrict__ out) {
    __shared__ float Xs[2][BM * LDS_STRIDE];   // 2 x 18 KB
    __shared__ float Ws[2][BN * LDS_STRIDE];   // 2 x 18 KB  (72 KB total)

    const int tid  = threadIdx.x;
    const int lane = tid & 31;
    const int wave = tid >> 5;
    const int wm   = wave & 1;   // 0..1 -> 64 rows
    const int wn   = wave >> 1;  // 0..3 -> 32 cols

    const size_t m0 = (size_t)blockIdx.x * BM;  // M = 16384 / 128 blocks
    const size_t n0 = (size_t)blockIdx.y * BN;  // N = 3072  / 128 blocks

    const int l16   = lane & 15;
    const int khalf = (lane >> 4) * 2;          // lanes 16-31 hold K+2,K+3

    const unsigned int xs_base = (unsigned int)(uintptr_t)&Xs[0][0];
    const unsigned int ws_base = (unsigned int)(uintptr_t)&Ws[0][0];
    const unsigned int buf_bytes = BM * LDS_STRIDE * 4;

    v8f acc[4][2];
    #pragma unroll
    for (int i = 0; i < 4; ++i)
        #pragma unroll
        for (int j = 0; j < 2; ++j)
            acc[i][j] = (v8f)0.0f;

    // Prologue: DMA first K-tile into buffer 0 (one wave issues; TDM ignores EXEC).
    if (wave == 0) {
        tdm_load_tile_f32(xs_base, x + m0 * D_MODEL, BK, BM, D_MODEL, 16384, D_MODEL);
        tdm_load_tile_f32(ws_base, w + n0 * D_MODEL, BK, BN, D_MODEL, N_OUT, D_MODEL);
        __builtin_amdgcn_s_wait_tensorcnt(0);
    }
    __syncthreads();

    const int KT = D_MODEL / BK;   // 32 K-tiles
    for (int kt = 0; kt < KT; ++kt) {
        const int cur = kt & 1;
        const int nxt = cur ^ 1;

        // Prefetch next K-tile via TDM while this one is consumed by WMMAs.
        if (wave == 0 && (kt + 1) < KT) {
            const int k0n = (kt + 1) * BK;
            tdm_load_tile_f32(xs_base + nxt * buf_bytes, x + m0 * D_MODEL + k0n,
                              BK, BM, D_MODEL, 16384, D_MODEL);
            tdm_load_tile_f32(ws_base + nxt * buf_bytes, w + n0 * D_MODEL + k0n,
                              BK, BN, D_MODEL, N_OUT, D_MODEL);
        }

        const float* __restrict__ xs = &Xs[cur][0];
        const float* __restrict__ wsp = &Ws[cur][0];

        #pragma unroll
        for (int kk = 0; kk < BK; kk += 4) {
            v2f a[4];
            #pragma unroll
            for (int i = 0; i < 4; ++i) {
                const float* p = &xs[(wm * 64 + i * 16 + l16) * LDS_STRIDE + kk + khalf];
                a[i].x = p[0];
                a[i].y = p[1];
            }
            v2f b[2];
            #pragma unroll
            for (int j = 0; j < 2; ++j) {
                const float* p = &wsp[(wn * 32 + j * 16 + l16) * LDS_STRIDE + kk + khalf];
                b[j].x = p[0];
                b[j].y = p[1];
            }
            #pragma unroll
            for (int i = 0; i < 4; ++i)
                #pragma unroll
                for (int j = 0; j < 2; ++j)
                    acc[i][j] = __builtin_amdgcn_wmma_f32_16x16x4_f32(
                        false, a[i], false, b[j], (short)0, acc[i][j], false, false);
        }

        // Wait for the prefetched tile (overlapped with the WMMAs above),
        // then make it visible to all waves.
        if (wave == 0) __builtin_amdgcn_s_wait_tensorcnt(0);
        __syncthreads();
    }

    // ------------------------------------------------------------------
    // LoRA up-projection via WMMA: acc += (T_sel * SCALING) @ B_sel^T.
    // Rank-16 contraction = 4 K-steps of 16x16x4. Each 128-wide N block is
    // entirely q / k / v, so this branch is uniform across the workgroup.
    // SCALING is folded into the A-fragment loads.
    // ------------------------------------------------------------------
    const float* __restrict__ bmat = nullptr;
    int t_off = 0;
    if (n0 < 1024) {                       // Q block
        bmat  = b_q;
        t_off = 0;
    } else if (n0 >= 2048) {               // V block (index by global n below)
        bmat  = b_v - (size_t)2048 * RANK;
        t_off = 16;
    }
    if (bmat) {
        #pragma unroll
        for (int kk = 0; kk < RANK; kk += 4) {
            v2f a[4];
            #pragma unroll
            for (int i = 0; i < 4; ++i) {
                const size_t m = m0 + wm * 64 + i * 16 + l16;
                const float* t = T + m * 32 + t_off + kk + khalf;
                a[i].x = t[0] * SCALING;
                a[i].y = t[1] * SCALING;
            }
            v2f b[2];
            #pragma unroll
            for (int j = 0; j < 2; ++j) {
                const size_t n = n0 + wn * 32 + j * 16 + l16;
                const float* p = bmat + n * RANK + kk + khalf;
                b[j].x = p[0];
                b[j].y = p[1];
            }
            #pragma unroll
            for (int i = 0; i < 4; ++i)
                #pragma unroll
                for (int j = 0; j < 2; ++j)
                    acc[i][j] = __builtin_amdgcn_wmma_f32_16x16x4_f32(
                        false, a[i], false, b[j], (short)0, acc[i][j], false, false);
        }
    }

    // Bias + store. C layout: VGPR r -> M = r + 8*(lane>=16), N = lane%16.
    // Streaming 192 MB output: nontemporal store so X (reused 24x) and
    // W (reused 128x) stay resident in the 192 MB L2.
    const int mhalf = (lane >> 4) * 8;
    #pragma unroll
    for (int j = 0; j < 2; ++j) {
        const size_t n  = n0 + wn * 32 + j * 16 + l16;
        const float  bn = bias[n];
        #pragma unroll
        for (int i = 0; i < 4; ++i) {
            #pragma unroll
            for (int r = 0; r < 8; ++r) {
                const size_t m = m0 + wm * 64 + i * 16 + mhalf + r;
                __builtin_nontemporal_store(acc[i][j][r] + bn,
                                            &out[m * (size_t)N_OUT + n]);
            }
        }
    }
}

// ---------------------------------------------------------------------------
extern "C" void kernel_launch(void* const* d_in, const int* in_sizes, int n_in,
                              void* d_out, int out_size, void* d_ws, size_t ws_size,
                              hipStream_t stream) {
    const float* x     = (const float*)d_in[0];
    const float* w_qkv = (const float*)d_in[1];
    const float* b_qkv = (const float*)d_in[2];
    const float* a_q   = (const float*)d_in[3];
    const float* b_q   = (const float*)d_in[4];
    const float* a_v   = (const float*)d_in[5];
    const float* b_v   = (const float*)d_in[6];
    float* out = (float*)d_out;
    float* T   = (float*)d_ws;   // 16384 x 32 floats = 2 MB scratch

    const int M = 4 * 4096;      // 16384 rows

    lora_down_kernel<<<M / 8, 256, 0, stream>>>(x, a_q, a_v, T);

    dim3 grid(M / BM, N_OUT / BN);   // 128 x 24
    qkv_wmma_kernel<<<grid, 256, 0, stream>>>(x, w_qkv, b_qkv, b_q, b_v, T, out);
}